// Actor_Critic_GAT_RNN_74938589381431
// MI455X (gfx1250) — compile-verified
//
#include <hip/hip_runtime.h>
#include <math.h>

#define NN 102400
#define EE 1638400
#define BB 1024

typedef __attribute__((ext_vector_type(2))) float v2f;
typedef __attribute__((ext_vector_type(8))) float v8f;

// ---- order-preserving float<->uint encoding for atomic max ----
static __device__ __forceinline__ unsigned ford(float f) {
  unsigned u = __float_as_uint(f);
  return (u & 0x80000000u) ? ~u : (u | 0x80000000u);
}
static __device__ __forceinline__ float funord(unsigned u) {
  return (u & 0x80000000u) ? __uint_as_float(u ^ 0x80000000u)
                           : __uint_as_float(~u);
}
#define ENC_NEGINF 0x007FFFFFu  // ford(-inf)

__global__ void k_fill_u32(unsigned* __restrict__ p, unsigned val, int n) {
  int i = blockIdx.x * blockDim.x + threadIdx.x;
  if (i < n) p[i] = val;
}

// ---- generic C[M,N] = act(A[M,K] @ W[N,K]^T + bias) via v_wmma_f32_16x16x4_f32
// one wave per 16x16 output tile; K multiple of 4, N multiple of 16, M multiple of 16
__global__ void k_wmma_linear(const float* __restrict__ A, const float* __restrict__ W,
                              const float* __restrict__ bias, float* __restrict__ C,
                              int M, int N, int K, int act) {
  int wave = (blockIdx.x * blockDim.x + threadIdx.x) >> 5;
  int lane = threadIdx.x & 31;
  int tiles_n = N >> 4;
  int tm = wave / tiles_n;
  int tn = wave - tm * tiles_n;
  if (tm >= (M >> 4)) return;           // whole-wave exit, EXEC stays full for WMMA
  int lrow = lane & 15;                  // M (for A) / N (for B) index within tile
  int lhi  = lane >> 4;                  // K-half selector
  const float* Arow = A + (size_t)(tm * 16 + lrow) * K;
  const float* Wrow = W + (size_t)(tn * 16 + lrow) * K;
  v8f c = {0.f, 0.f, 0.f, 0.f, 0.f, 0.f, 0.f, 0.f};
  for (int kk = 0; kk < K; kk += 4) {
    int k = kk + 2 * lhi;
    v2f a, b;
    a[0] = Arow[k];     a[1] = Arow[k + 1];   // A 16x4: lanes0-15 K={0,1}, lanes16-31 K={2,3}
    b[0] = Wrow[k];     b[1] = Wrow[k + 1];   // B 4x16 mirrored layout
    c = __builtin_amdgcn_wmma_f32_16x16x4_f32(false, a, false, b, (short)0, c, false, false);
  }
  int col = tn * 16 + lrow;
  float bv = bias ? bias[col] : 0.f;
#pragma unroll
  for (int r = 0; r < 8; ++r) {
    int row = tm * 16 + r + lhi * 8;     // C layout: VGPR r -> M=r (lanes0-15) / r+8 (lanes16-31)
    float v = c[r] + bv;
    if (act) v = tanhf(v);
    C[(size_t)row * N + col] = v;
  }
}

// per-node attention coefficients: as[n,h] = sum_k h[n,h,k]*a_src[h,k]
__global__ void k_attn(const float* __restrict__ h, const float* __restrict__ avs,
                       const float* __restrict__ avd, float* __restrict__ os,
                       float* __restrict__ od, int n, int H, int D) {
  int i = blockIdx.x * blockDim.x + threadIdx.x;
  if (i >= n * H) return;
  int node = i / H, hd = i - node * H;
  const float* hp = h + (size_t)node * H * D + hd * D;
  float s = 0.f, d = 0.f;
  for (int k = 0; k < D; ++k) { float v = hp[k]; s += v * avs[hd * D + k]; d += v * avd[hd * D + k]; }
  os[i] = s; od[i] = d;
}

__global__ void k_edge_max(const int* __restrict__ src, const int* __restrict__ dst,
                           const float* __restrict__ as_, const float* __restrict__ ad_,
                           unsigned* __restrict__ m, int E, int n, int H) {
  long long i = (long long)blockIdx.x * blockDim.x + threadIdx.x;
  long long total = (long long)(E + n) * H;
  if (i >= total) return;
  int e = (int)(i / H), hd = (int)(i - (long long)e * H);
  int s, d;
  if (e < E) { s = src[e]; d = dst[e]; } else { s = d = e - E; }  // self-loops appended
  float x = as_[s * H + hd] + ad_[d * H + hd];
  x = x > 0.f ? x : 0.2f * x;                                     // leaky_relu(0.2)
  atomicMax(&m[d * H + hd], ford(x));
}

__global__ void k_edge_accum(const int* __restrict__ src, const int* __restrict__ dst,
                             const float* __restrict__ as_, const float* __restrict__ ad_,
                             const unsigned* __restrict__ m, float* __restrict__ den,
                             float* __restrict__ num, const float* __restrict__ hfeat,
                             int E, int n, int H, int D) {
  long long i = (long long)blockIdx.x * blockDim.x + threadIdx.x;
  long long total = (long long)(E + n) * H;
  if (i >= total) return;
  int e = (int)(i / H), hd = (int)(i - (long long)e * H);
  int s, d;
  if (e < E) { s = src[e]; d = dst[e]; } else { s = d = e - E; }
  float x = as_[s * H + hd] + ad_[d * H + hd];
  x = x > 0.f ? x : 0.2f * x;
  float p = __expf(x - funord(m[d * H + hd]));
  atomicAdd(&den[d * H + hd], p);
  const float* hs = hfeat + (size_t)s * H * D + hd * D;
  float* nd = num + (size_t)d * H * D + hd * D;
  for (int k = 0; k < D; ++k) atomicAdd(&nd[k], hs[k] * p);
}

// layer1 epilogue: h1 = tanh(num/den + bias)  (in-place overwrite of h1)
__global__ void k_finalize1(const float* __restrict__ num, const float* __restrict__ den,
                            const float* __restrict__ bias, float* __restrict__ h, int n) {
  int i = blockIdx.x * blockDim.x + threadIdx.x;
  if (i >= n * 96) return;
  int node = i / 96, col = i - node * 96, hd = col >> 4;
  h[i] = tanhf(num[i] / den[node * 6 + hd] + bias[col]);
}

// gather robot rows of layer-2 output and concat robot_features -> fcin [B,36]
__global__ void k_fcin(const float* __restrict__ num2, const float* __restrict__ den2,
                       const float* __restrict__ c2b, const int* __restrict__ ridx,
                       const float* __restrict__ rfeat, float* __restrict__ fcin) {
  int i = blockIdx.x * blockDim.x + threadIdx.x;
  if (i >= BB * 36) return;
  int b = i / 36, c = i - b * 36;
  if (c < 32) { int nd = ridx[b]; fcin[i] = num2[(size_t)nd * 32 + c] / den2[nd] + c2b[c]; }
  else fcin[i] = rfeat[b * 4 + (c - 32)];
}

// sequential GRU over 1024 steps, single workgroup, whh^T resident in LDS
__global__ void __launch_bounds__(192) k_gru(const float* __restrict__ gi,
                                             const float* __restrict__ whh,
                                             const float* __restrict__ bhh,
                                             float* __restrict__ ys) {
  __shared__ float whhT[64 * 192];   // [k][j] conflict-free broadcast reads
  __shared__ float hs[64];
  __shared__ float gh[192];
  int j = threadIdx.x;
  for (int idx = j; idx < 64 * 192; idx += 192) {
    int k = idx / 192, row = idx - k * 192;
    whhT[idx] = whh[row * 64 + k];
  }
  if (j < 64) hs[j] = 0.f;
  float bj = bhh[j];
  __syncthreads();
  for (int t = 0; t < BB; ++t) {
    float acc = bj;
#pragma unroll 8
    for (int k = 0; k < 64; ++k) acc += whhT[k * 192 + j] * hs[k];
    gh[j] = acc;
    __syncthreads();
    float hn = 0.f;
    if (j < 64) {
      const float* g = gi + (size_t)t * 192;
      float r  = 1.f / (1.f + __expf(-(g[j]       + gh[j])));
      float z  = 1.f / (1.f + __expf(-(g[64 + j]  + gh[64 + j])));
      float ng = tanhf(g[128 + j] + r * gh[128 + j]);
      hn = (1.f - z) * ng + z * hs[j];
      ys[(size_t)t * 64 + j] = hn;
    }
    __syncthreads();
    if (j < 64) hs[j] = hn;
    __syncthreads();
  }
}

__global__ void k_fc2(const float* __restrict__ ys, const float* __restrict__ W,
                      const float* __restrict__ b, float* __restrict__ out) {
  int i = blockIdx.x * blockDim.x + threadIdx.x;
  if (i >= BB * 11) return;
  int bt = i / 11, o = i - bt * 11;
  float acc = b[o];
  const float* y = ys + (size_t)bt * 64;
  for (int k = 0; k < 64; ++k) acc += y[k] * W[o * 64 + k];
  out[i] = acc;
}

static inline int cdiv(long long a, long long b) { return (int)((a + b - 1) / b); }

extern "C" void kernel_launch(void* const* d_in, const int* in_sizes, int n_in,
                              void* d_out, int out_size, void* d_ws, size_t ws_size,
                              hipStream_t stream) {
  const float* x     = (const float*)d_in[0];
  const int*   ei    = (const int*)d_in[1];
  const int*   ridx  = (const int*)d_in[2];
  const float* rfeat = (const float*)d_in[3];
  const float* c1_W  = (const float*)d_in[4];
  const float* c1_as = (const float*)d_in[5];
  const float* c1_ad = (const float*)d_in[6];
  const float* c1_b  = (const float*)d_in[7];
  const float* c2_W  = (const float*)d_in[8];
  const float* c2_as = (const float*)d_in[9];
  const float* c2_ad = (const float*)d_in[10];
  const float* c2_b  = (const float*)d_in[11];
  const float* fc1_W = (const float*)d_in[12];
  const float* fc1_b = (const float*)d_in[13];
  const float* wih   = (const float*)d_in[14];
  const float* whh   = (const float*)d_in[15];
  const float* bih   = (const float*)d_in[16];
  const float* bhh   = (const float*)d_in[17];
  const float* fc2_W = (const float*)d_in[18];
  const float* fc2_b = (const float*)d_in[19];
  const int* src = ei;
  const int* dst = ei + EE;

  float* ws = (float*)d_ws;
  size_t off = 0;
  float* h1  = ws + off; off += (size_t)NN * 96;
  float* as1 = ws + off; off += (size_t)NN * 6;
  float* ad1 = ws + off; off += (size_t)NN * 6;
  float* h2  = ws + off; off += (size_t)NN * 32;
  float* as2 = ws + off; off += (size_t)NN;
  float* ad2 = ws + off; off += (size_t)NN;
  float* zero_base = ws + off;               // contiguous zero-init region:
  float* num1 = ws + off; off += (size_t)NN * 96;
  float* den1 = ws + off; off += (size_t)NN * 6;
  float* num2 = ws + off; off += (size_t)NN * 32;
  float* den2 = ws + off; off += (size_t)NN;
  size_t zero_cnt = (size_t)NN * (96 + 6 + 32 + 1);
  unsigned* m1 = (unsigned*)(ws + off); off += (size_t)NN * 6;   // contiguous -inf region
  unsigned* m2 = (unsigned*)(ws + off); off += (size_t)NN;
  float* fcin  = ws + off; off += (size_t)BB * 36;
  float* fc1o  = ws + off; off += (size_t)BB * 64;
  float* gi    = ws + off; off += (size_t)BB * 192;
  float* ysb   = ws + off; off += (size_t)BB * 64;

  const int T = 256;
  // init accumulators (d_ws is poisoned)
  k_fill_u32<<<cdiv(zero_cnt, T), T, 0, stream>>>((unsigned*)zero_base, 0u, (int)zero_cnt);
  k_fill_u32<<<cdiv((size_t)NN * 7, T), T, 0, stream>>>(m1, ENC_NEGINF, NN * 7);

  // ---- GAT layer 1 ----
  long long w1 = (long long)(NN / 16) * (96 / 16) * 32;
  k_wmma_linear<<<cdiv(w1, T), T, 0, stream>>>(x, c1_W, nullptr, h1, NN, 96, 4, 0);
  k_attn<<<cdiv((long long)NN * 6, T), T, 0, stream>>>(h1, c1_as, c1_ad, as1, ad1, NN, 6, 16);
  long long et1 = (long long)(EE + NN) * 6;
  k_edge_max<<<cdiv(et1, T), T, 0, stream>>>(src, dst, as1, ad1, m1, EE, NN, 6);
  k_edge_accum<<<cdiv(et1, T), T, 0, stream>>>(src, dst, as1, ad1, m1, den1, num1, h1, EE, NN, 6, 16);
  k_finalize1<<<cdiv((long long)NN * 96, T), T, 0, stream>>>(num1, den1, c1_b, h1, NN);

  // ---- GAT layer 2 ----
  long long w2 = (long long)(NN / 16) * (32 / 16) * 32;
  k_wmma_linear<<<cdiv(w2, T), T, 0, stream>>>(h1, c2_W, nullptr, h2, NN, 32, 96, 0);
  k_attn<<<cdiv((long long)NN, T), T, 0, stream>>>(h2, c2_as, c2_ad, as2, ad2, NN, 1, 32);
  long long et2 = (long long)(EE + NN);
  k_edge_max<<<cdiv(et2, T), T, 0, stream>>>(src, dst, as2, ad2, m2, EE, NN, 1);
  k_edge_accum<<<cdiv(et2, T), T, 0, stream>>>(src, dst, as2, ad2, m2, den2, num2, h2, EE, NN, 1, 32);

  // ---- gather + fc1 + GRU input precompute ----
  k_fcin<<<cdiv((long long)BB * 36, T), T, 0, stream>>>(num2, den2, c2_b, ridx, rfeat, fcin);
  k_wmma_linear<<<cdiv((long long)(BB / 16) * 4 * 32, T), T, 0, stream>>>(fcin, fc1_W, fc1_b, fc1o, BB, 64, 36, 1);
  k_wmma_linear<<<cdiv((long long)(BB / 16) * 12 * 32, T), T, 0, stream>>>(fc1o, wih, bih, gi, BB, 192, 64, 0);

  // ---- sequential GRU + fc2 ----
  k_gru<<<1, 192, 0, stream>>>(gi, whh, bhh, ysb);
  k_fc2<<<cdiv((long long)BB * 11, T), T, 0, stream>>>(ysb, fc2_W, fc2_b, (float*)d_out);
}